// Criterion_87978110091394
// MI455X (gfx1250) — compile-verified
//
#include <hip/hip_runtime.h>
#include <math.h>

typedef __attribute__((ext_vector_type(2))) float v2f;
typedef __attribute__((ext_vector_type(8))) float v8f;

#define NROWS      8192
#define DIM        64
#define NLAB       80
#define EPS_POS    (1.0f - 1e-5f)
#define MARGIN_C   0.1f
#define POS_SCALE  2.0f
#define NEG_SCALE  40.0f
#define NTILES     (NROWS / 16)   // 512 column tiles
#define LDA        68             // padded LDS stride (bank-spread, keeps 8B align)

// ---- monotonic float<->uint mapping for atomic min/max on shared memory ----
__device__ __forceinline__ unsigned f2ord(float f) {
  unsigned u = __float_as_uint(f);
  return (u & 0x80000000u) ? ~u : (u | 0x80000000u);
}
__device__ __forceinline__ float ord2f(unsigned u) {
  unsigned b = (u & 0x80000000u) ? (u ^ 0x80000000u) : ~u;
  return __uint_as_float(b);
}

// ---- pack 80 label ints -> 3 uint32 bitmask words (4th word pad) ----
__global__ void pack_labels_kernel(const int* __restrict__ labels,
                                   unsigned* __restrict__ masks) {
  int row = blockIdx.x * blockDim.x + threadIdx.x;
  if (row >= NROWS) return;
  const int* lr = labels + (size_t)row * NLAB;
  unsigned w0 = 0, w1 = 0, w2 = 0;
  #pragma unroll
  for (int t = 0; t < 32; ++t) w0 |= (lr[t]      > 0) ? (1u << t) : 0u;
  #pragma unroll
  for (int t = 0; t < 32; ++t) w1 |= (lr[32 + t] > 0) ? (1u << t) : 0u;
  #pragma unroll
  for (int t = 0; t < 16; ++t) w2 |= (lr[64 + t] > 0) ? (1u << t) : 0u;
  masks[row * 4 + 0] = w0;
  masks[row * 4 + 1] = w1;
  masks[row * 4 + 2] = w2;
  masks[row * 4 + 3] = 0u;
}

// Compute one 16x16 f32 sim tile with 16 chained V_WMMA_F32_16X16X4_F32.
// afrag: per-k-step A fragments (persistent). colBase already includes the
// lane's column row pointer + half*2 k-offset, so each k-step is one aligned
// 8-byte load (B 4x16 layout: VGPR0 = rows K0/K2 across lane halves, VGPR1 = K1/K3).
__device__ __forceinline__ v8f tile_wmma(const v2f* afrag, const float* colBase) {
  v8f c = {0.f, 0.f, 0.f, 0.f, 0.f, 0.f, 0.f, 0.f};
  #pragma unroll
  for (int kk = 0; kk < 16; ++kk) {
    v2f b = *(const v2f*)(colBase + kk * 4);
    c = __builtin_amdgcn_wmma_f32_16x16x4_f32(false, afrag[kk], false, b,
                                              (short)0, c, false, false);
  }
  return c;
}

__global__ __launch_bounds__(256) void ms_loss_kernel(
    const float* __restrict__ feats, const unsigned* __restrict__ masks,
    float* __restrict__ out) {
  __shared__ float    ldsA[16 * LDA];
  __shared__ unsigned rowMask[16][3];
  __shared__ float    normAcc[16];
  __shared__ unsigned minPosU[16];
  __shared__ unsigned maxNegU[16];
  __shared__ float    posSumS[16];
  __shared__ float    negSumS[16];
  __shared__ float    rnormS[16];
  __shared__ float    minPosS[16];
  __shared__ float    maxNegS[16];
  __shared__ float    lossS[16];

  const int tid  = threadIdx.x;
  const int wave = tid >> 5;          // 8 waves per workgroup (wave32)
  const int lane = tid & 31;
  const int half = lane >> 4;         // lane-half split per WMMA layout
  const int l16  = lane & 15;
  const int r0   = blockIdx.x * 16;   // this block's 16 rows

  if (tid < 16) {
    normAcc[tid] = 0.f;
    minPosU[tid] = f2ord(__builtin_inff());
    maxNegU[tid] = f2ord(-__builtin_inff());
    posSumS[tid] = 0.f;
    negSumS[tid] = 0.f;
  }
  // Stage the 16x64 A rows into LDS (padded stride).
  for (int idx = tid; idx < 16 * DIM; idx += 256) {
    int r = idx >> 6, cc = idx & 63;
    ldsA[r * LDA + cc] = feats[(size_t)(r0 + r) * DIM + cc];
  }
  if (tid < 48) rowMask[tid / 3][tid % 3] = masks[(size_t)(r0 + tid / 3) * 4 + (tid % 3)];
  __syncthreads();

  // Persistent A fragments: A 16x4 f32 layout (lanes 0-15: K0/K1, lanes 16-31: K2/K3).
  v2f afrag[16];
  #pragma unroll
  for (int kk = 0; kk < 16; ++kk) {
    int ks = kk * 4 + half * 2;
    afrag[kk].x = ldsA[l16 * LDA + ks];
    afrag[kk].y = ldsA[l16 * LDA + ks + 1];
  }

  // ---------------- sweep 1: row norms of the Gram rows ----------------
  {
    float sq[8] = {0.f, 0.f, 0.f, 0.f, 0.f, 0.f, 0.f, 0.f};
    for (int t = wave; t < NTILES; t += 8) {
      const int c0 = t * 16;
      const float* colBase = feats + (size_t)(c0 + l16) * DIM + half * 2;
      v8f c = tile_wmma(afrag, colBase);
      #pragma unroll
      for (int v = 0; v < 8; ++v) sq[v] += c[v] * c[v];
    }
    #pragma unroll
    for (int v = 0; v < 8; ++v) {
      float s = sq[v];
      #pragma unroll
      for (int m = 1; m < 16; m <<= 1) s += __shfl_xor(s, m, 32);
      if (l16 == 0) atomicAdd(&normAcc[v + half * 8], s);
    }
  }
  __syncthreads();
  if (tid < 16) rnormS[tid] = 1.0f / fmaxf(sqrtf(normAcc[tid]), 1e-12f);
  __syncthreads();

  // ---------------- sweep 2: min positive / max negative ----------------
  {
    float mp[8], mn[8];
    #pragma unroll
    for (int v = 0; v < 8; ++v) { mp[v] = __builtin_inff(); mn[v] = -__builtin_inff(); }
    for (int t = wave; t < NTILES; t += 8) {
      const int c0 = t * 16;
      const float* colBase = feats + (size_t)(c0 + l16) * DIM + half * 2;
      v8f c = tile_wmma(afrag, colBase);
      const unsigned* cm = masks + (size_t)(c0 + l16) * 4;
      unsigned m0 = cm[0], m1 = cm[1], m2 = cm[2];
      #pragma unroll
      for (int v = 0; v < 8; ++v) {
        int row = v + half * 8;
        bool pos = ((rowMask[row][0] & m0) | (rowMask[row][1] & m1) |
                    (rowMask[row][2] & m2)) != 0u;
        float sim = c[v] * rnormS[row];
        if (pos) {
          if (sim < EPS_POS) mp[v] = fminf(mp[v], sim);
        } else {
          mn[v] = fmaxf(mn[v], sim);
        }
      }
    }
    #pragma unroll
    for (int v = 0; v < 8; ++v) {
      float a = mp[v], b = mn[v];
      #pragma unroll
      for (int m = 1; m < 16; m <<= 1) {
        a = fminf(a, __shfl_xor(a, m, 32));
        b = fmaxf(b, __shfl_xor(b, m, 32));
      }
      if (l16 == 0) {
        atomicMin(&minPosU[v + half * 8], f2ord(a));
        atomicMax(&maxNegU[v + half * 8], f2ord(b));
      }
    }
  }
  __syncthreads();
  if (tid < 16) {
    minPosS[tid] = ord2f(minPosU[tid]);
    maxNegS[tid] = ord2f(maxNegU[tid]);
  }
  __syncthreads();

  // ---------------- sweep 3: mined-pair exp sums ----------------
  {
    float ps[8] = {0.f, 0.f, 0.f, 0.f, 0.f, 0.f, 0.f, 0.f};
    float ns[8] = {0.f, 0.f, 0.f, 0.f, 0.f, 0.f, 0.f, 0.f};
    for (int t = wave; t < NTILES; t += 8) {
      const int c0 = t * 16;
      const float* colBase = feats + (size_t)(c0 + l16) * DIM + half * 2;
      v8f c = tile_wmma(afrag, colBase);
      const unsigned* cm = masks + (size_t)(c0 + l16) * 4;
      unsigned m0 = cm[0], m1 = cm[1], m2 = cm[2];
      #pragma unroll
      for (int v = 0; v < 8; ++v) {
        int row = v + half * 8;
        bool pos = ((rowMask[row][0] & m0) | (rowMask[row][1] & m1) |
                    (rowMask[row][2] & m2)) != 0u;
        float sim = c[v] * rnormS[row];
        if (pos) {
          if (sim < EPS_POS && (sim - MARGIN_C) < maxNegS[row])
            ps[v] += __expf(-POS_SCALE * (sim - MARGIN_C));
        } else {
          if ((sim + MARGIN_C) > minPosS[row])
            ns[v] += __expf(NEG_SCALE * (sim - MARGIN_C));
        }
      }
    }
    #pragma unroll
    for (int v = 0; v < 8; ++v) {
      float a = ps[v], b = ns[v];
      #pragma unroll
      for (int m = 1; m < 16; m <<= 1) {
        a += __shfl_xor(a, m, 32);
        b += __shfl_xor(b, m, 32);
      }
      if (l16 == 0) {
        atomicAdd(&posSumS[v + half * 8], a);
        atomicAdd(&negSumS[v + half * 8], b);
      }
    }
  }
  __syncthreads();

  // ---------------- finalize 16 rows, one atomic per block ----------------
  if (tid < 16) {
    float psv = posSumS[tid], nsv = negSumS[tid];
    bool valid = (minPosS[tid] < __builtin_inff()) &&
                 (maxNegS[tid] > -__builtin_inff()) &&
                 (psv > 0.f) && (nsv > 0.f);
    lossS[tid] = valid ? (0.5f * log1pf(psv) + 0.025f * log1pf(nsv)) : 0.f;
  }
  __syncthreads();
  if (tid == 0) {
    float s = 0.f;
    #pragma unroll
    for (int i = 0; i < 16; ++i) s += lossS[i];
    atomicAdd(out, s * (1.0f / (float)NROWS));
  }
}

extern "C" void kernel_launch(void* const* d_in, const int* in_sizes, int n_in,
                              void* d_out, int out_size, void* d_ws, size_t ws_size,
                              hipStream_t stream) {
  const float* feats  = (const float*)d_in[0];
  const int*   labels = (const int*)d_in[1];
  unsigned*    masks  = (unsigned*)d_ws;   // 8192 * 4 uint32 = 128 KB
  float*       out    = (float*)d_out;

  hipMemsetAsync(d_out, 0, sizeof(float), stream);
  pack_labels_kernel<<<(NROWS + 255) / 256, 256, 0, stream>>>(labels, masks);
  ms_loss_kernel<<<NROWS / 16, 256, 0, stream>>>(feats, masks, out);
}